// HMMN_29884382445751
// MI455X (gfx1250) — compile-verified
//
#include <hip/hip_runtime.h>

// Problem constants (from reference setup_inputs)
#define B_   4
#define De_  128
#define Do_  512
#define T_   10
#define H_   32
#define W_   32
#define THW_ (T_ * H_ * W_)   // 10240
#define HW_  (H_ * W_)        // 1024

typedef __attribute__((ext_vector_type(2))) float v2f;
typedef __attribute__((ext_vector_type(8))) float v8f;

__device__ __constant__ const float kScale  = 0.08838834764831845f; // 1/sqrt(128)
__device__ __constant__ const float kInv2S2 = 1.0f / 18.0f;         // 1/(2*sigma^2), sigma=3

// ---------------------------------------------------------------------------
// Kernel 1: p[b, t, q] = sum_k mi[b,k,t] * qi[b,k,q] / sqrt(De)
// One wave computes a 16(M=t) x 64(N=q) tile via V_WMMA_F32_16X16X4_F32.
// A layout (16x4 f32): lane = M%16, half-wave picks K pair {0,1} vs {2,3}.
// ---------------------------------------------------------------------------
__global__ __launch_bounds__(256) void k_gemm_p(const float* __restrict__ mi,
                                                const float* __restrict__ qi,
                                                float* __restrict__ p)
{
    const int lane  = threadIdx.x & 31;
    const int wave  = blockIdx.x * 8 + (threadIdx.x >> 5);
    const int mlane = lane & 15;
    const int khalf = lane >> 4;

    const int ntiles = HW_ / 64;          // 16
    const int mtiles = THW_ / 16;         // 640
    const int perB   = ntiles * mtiles;   // 10240 tiles per batch
    const int b  = wave / perB;
    const int r  = wave % perB;
    const int m0 = (r / ntiles) * 16;
    const int n0 = (r % ntiles) * 64;

    const float* A  = mi + (size_t)b * De_ * THW_;   // A[k*THW + t]
    const float* Bq = qi + (size_t)b * De_ * HW_;    // B[k*HW + q]

    v8f acc0 = {}, acc1 = {}, acc2 = {}, acc3 = {};

    for (int k0 = 0; k0 < De_; k0 += 4) {
        const int ka = k0 + khalf * 2;
        v2f a;
        a.x = A[(size_t)(ka + 0) * THW_ + m0 + mlane];
        a.y = A[(size_t)(ka + 1) * THW_ + m0 + mlane];

        const float* Bk0 = Bq + (size_t)(ka + 0) * HW_ + n0 + mlane;
        const float* Bk1 = Bk0 + HW_;
        v2f b0, b1, b2, b3;
        b0.x = Bk0[0];  b0.y = Bk1[0];
        b1.x = Bk0[16]; b1.y = Bk1[16];
        b2.x = Bk0[32]; b2.y = Bk1[32];
        b3.x = Bk0[48]; b3.y = Bk1[48];

        acc0 = __builtin_amdgcn_wmma_f32_16x16x4_f32(false, a, false, b0, (short)0, acc0, false, false);
        acc1 = __builtin_amdgcn_wmma_f32_16x16x4_f32(false, a, false, b1, (short)0, acc1, false, false);
        acc2 = __builtin_amdgcn_wmma_f32_16x16x4_f32(false, a, false, b2, (short)0, acc2, false, false);
        acc3 = __builtin_amdgcn_wmma_f32_16x16x4_f32(false, a, false, b3, (short)0, acc3, false, false);
    }

    // C/D layout: VGPR r -> M = m0 + r (+8 for upper half-wave), N = n-lane
    float* prow = p + (size_t)b * THW_ * HW_;
#pragma unroll
    for (int r8 = 0; r8 < 8; ++r8) {
        const int m = m0 + r8 + khalf * 8;
        float* dst = prow + (size_t)m * HW_ + n0 + mlane;
        dst[0]  = acc0[r8] * kScale;
        dst[16] = acc1[r8] * kScale;
        dst[32] = acc2[r8] * kScale;
        dst[48] = acc3[r8] * kScale;
    }
}

// ---------------------------------------------------------------------------
// Kernel 2: row argmax over the HW query grid (first-occurrence tie-break,
// matching jnp.argmax). One wave per row, 40960 rows.
// ---------------------------------------------------------------------------
__global__ __launch_bounds__(256) void k_rowargmax(const float* __restrict__ p,
                                                   int* __restrict__ idx)
{
    const int lane = threadIdx.x & 31;
    const int row  = blockIdx.x * 8 + (threadIdx.x >> 5);
    const float* pr = p + (size_t)row * HW_;

    float best = -3.402823466e38f;
    int   bi   = 0;
    for (int q = lane; q < HW_; q += 32) {
        const float v = pr[q];
        if (v > best) { best = v; bi = q; }
    }
#pragma unroll
    for (int off = 16; off > 0; off >>= 1) {
        const float ov = __shfl_xor(best, off, 32);
        const int   oi = __shfl_xor(bi,   off, 32);
        if (ov > best || (ov == best && oi < bi)) { best = ov; bi = oi; }
    }
    if (lane == 0) idx[row] = bi;
}

// ---------------------------------------------------------------------------
// Kernel 3: per query column q (over the t = THW axis):
//   p_w[t,q] = exp(p[t,q]-max)*g(t,q) / sum_t exp(p-max)*g    (in place)
// Block = 32 columns x 8 t-slices; coalesced loads; LDS cross-wave reduce.
// ---------------------------------------------------------------------------
__global__ __launch_bounds__(256) void k_colpass(float* __restrict__ p,
                                                 const int* __restrict__ idx)
{
    __shared__ float red[8][32];
    const int lane = threadIdx.x & 31;
    const int ws   = threadIdx.x >> 5;
    const int b    = blockIdx.x >> 5;          // 32 blocks per batch
    const int q    = ((blockIdx.x & 31) << 5) + lane;

    float* pc      = p + (size_t)b * THW_ * HW_ + q;
    const int* ib  = idx + b * THW_;

    // pass 1: column max
    float mx = -3.402823466e38f;
    for (int t = ws; t < THW_; t += 8) mx = fmaxf(mx, pc[(size_t)t * HW_]);
    red[ws][lane] = mx;
    __syncthreads();
    if (ws == 0) {
        float m = red[0][lane];
#pragma unroll
        for (int i = 1; i < 8; ++i) m = fmaxf(m, red[i][lane]);
        red[0][lane] = m;
    }
    __syncthreads();
    mx = red[0][lane];
    __syncthreads();

    // pass 2: e = exp(p - max) * gauss(idx[t], q); store in place; accumulate
    const float qy = (float)(q >> 5);
    const float qx = (float)(q & 31);
    float s = 0.0f;
    for (int t = ws; t < THW_; t += 8) {
        const int   id = ib[t];
        const float dy = qy - (float)(id >> 5);
        const float dx = qx - (float)(id & 31);
        const float g  = __expf(-(dx * dx + dy * dy) * kInv2S2);
        const float e  = __expf(pc[(size_t)t * HW_] - mx) * g;
        pc[(size_t)t * HW_] = e;
        s += e;
    }
    red[ws][lane] = s;
    __syncthreads();
    if (ws == 0) {
        float tsum = red[0][lane];
#pragma unroll
        for (int i = 1; i < 8; ++i) tsum += red[i][lane];
        red[0][lane] = tsum;
    }
    __syncthreads();
    const float inv = 1.0f / red[0][lane];

    // pass 3: scale own elements (same thread wrote them -> ordered per-thread)
    for (int t = ws; t < THW_; t += 8) pc[(size_t)t * HW_] *= inv;
}

// ---------------------------------------------------------------------------
// Kernel 4: mem[b, d, q] = sum_t mo[b,d,t] * p_w[b,t,q]  (K = 10240)
// One wave -> 16x64 tile; p_w stays L2-resident (168 MB < 192 MB L2).
// ---------------------------------------------------------------------------
__global__ __launch_bounds__(256) void k_gemm_mem(const float* __restrict__ mo,
                                                  const float* __restrict__ pw,
                                                  float* __restrict__ out)
{
    const int lane  = threadIdx.x & 31;
    const int wave  = blockIdx.x * 8 + (threadIdx.x >> 5);
    const int mlane = lane & 15;
    const int khalf = lane >> 4;

    const int ntiles = HW_ / 64;          // 16
    const int perB   = (Do_ / 16) * ntiles; // 512
    const int b  = wave / perB;
    const int r  = wave % perB;
    const int m0 = (r / ntiles) * 16;
    const int n0 = (r % ntiles) * 64;

    const float* arow = mo + (size_t)b * Do_ * THW_ + (size_t)(m0 + mlane) * THW_;
    const float* Bp   = pw + (size_t)b * THW_ * HW_;

    v8f acc0 = {}, acc1 = {}, acc2 = {}, acc3 = {};

    for (int k0 = 0; k0 < THW_; k0 += 4) {
        const int ka = k0 + khalf * 2;
        const v2f a = *(const v2f*)(arow + ka);           // contiguous K pair

        const float* Bk0 = Bp + (size_t)(ka + 0) * HW_ + n0 + mlane;
        const float* Bk1 = Bk0 + HW_;
        __builtin_prefetch(Bk0 + (size_t)32 * HW_, 0, 3); // global_prefetch_b8

        v2f b0, b1, b2, b3;
        b0.x = Bk0[0];  b0.y = Bk1[0];
        b1.x = Bk0[16]; b1.y = Bk1[16];
        b2.x = Bk0[32]; b2.y = Bk1[32];
        b3.x = Bk0[48]; b3.y = Bk1[48];

        acc0 = __builtin_amdgcn_wmma_f32_16x16x4_f32(false, a, false, b0, (short)0, acc0, false, false);
        acc1 = __builtin_amdgcn_wmma_f32_16x16x4_f32(false, a, false, b1, (short)0, acc1, false, false);
        acc2 = __builtin_amdgcn_wmma_f32_16x16x4_f32(false, a, false, b2, (short)0, acc2, false, false);
        acc3 = __builtin_amdgcn_wmma_f32_16x16x4_f32(false, a, false, b3, (short)0, acc3, false, false);
    }

    float* obase = out + (size_t)b * (2 * Do_) * HW_;     // channels [0,512) of concat
#pragma unroll
    for (int r8 = 0; r8 < 8; ++r8) {
        const int m = m0 + r8 + khalf * 8;
        float* dst = obase + (size_t)m * HW_ + n0 + mlane;
        dst[0]  = acc0[r8];
        dst[16] = acc1[r8];
        dst[32] = acc2[r8];
        dst[48] = acc3[r8];
    }
}

// ---------------------------------------------------------------------------
// Kernel 5: concat — copy q_out into channels [512, 1024) of mem_out.
// ---------------------------------------------------------------------------
__global__ __launch_bounds__(256) void k_copy_qout(const float* __restrict__ qo,
                                                   float* __restrict__ out)
{
    const int i   = blockIdx.x * 256 + threadIdx.x;       // 0 .. B*Do*HW-1
    const int per = Do_ * HW_;
    const int b   = i / per;
    const int rem = i - b * per;                          // c*HW + q
    out[(size_t)b * (2 * Do_) * HW_ + (size_t)Do_ * HW_ + rem] = qo[i];
}

// ---------------------------------------------------------------------------
extern "C" void kernel_launch(void* const* d_in, const int* in_sizes, int n_in,
                              void* d_out, int out_size, void* d_ws, size_t ws_size,
                              hipStream_t stream)
{
    const float* m_in  = (const float*)d_in[0];
    const float* m_out = (const float*)d_in[1];
    const float* q_in  = (const float*)d_in[2];
    const float* q_out = (const float*)d_in[3];

    float* out = (float*)d_out;
    float* pw  = out + (size_t)B_ * (2 * Do_) * HW_;  // p / p_w output region
    int*   idx = (int*)d_ws;                          // 40960 ints

    // 1) p = mi^T qi / sqrt(De)  -> written into p_w region
    k_gemm_p<<<(B_ * (THW_ / 16) * (HW_ / 64)) / 8, 256, 0, stream>>>(m_in, q_in, pw);
    // 2) row argmax over query grid
    k_rowargmax<<<(B_ * THW_) / 8, 256, 0, stream>>>(pw, idx);
    // 3) fused softmax * gaussian + renorm, in place (p -> p_w)
    k_colpass<<<B_ * (HW_ / 32), 256, 0, stream>>>(pw, idx);
    // 4) mem = mo * p_w  -> channels [0,512) of mem_out
    k_gemm_mem<<<(B_ * (Do_ / 16) * (HW_ / 64)) / 8, 256, 0, stream>>>(m_out, pw, out);
    // 5) concat q_out -> channels [512,1024)
    k_copy_qout<<<(B_ * Do_ * HW_) / 256, 256, 0, stream>>>(q_out, out);
}